// EndPointSpline_18124761989444
// MI455X (gfx1250) — compile-verified
//
#include <hip/hip_runtime.h>
#include <stdint.h>

// Problem constants (match the reference)
#define QN 2048
#define BN 128
#define TN 128
#define DN 512

typedef float v4f __attribute__((ext_vector_type(4)));

// One block (128 threads = 4 waves) produces one output row out[b, q, 0:512].
// q is the fast-varying block index -> all blocks for a given b reuse the same
// ~258KB knot slab (L2-resident), while the 512MB output streams out with NT
// stores. The two gathered knot rows are staged through LDS with the gfx1250
// async global->LDS path (ASYNCcnt).
__global__ __launch_bounds__(128, 8)
void EndPointSpline_18124761989444_kernel(const float* __restrict__ query_t,
                                          const float* __restrict__ knots,   // [B, T-2, D]
                                          const float* __restrict__ x0,      // [1, B, D]
                                          const float* __restrict__ x1,      // [1, B, D]
                                          const float* __restrict__ tgrid,   // [T, B]
                                          float* __restrict__ out)           // [B, Q, D]
{
    __shared__ __align__(16) float smem[2 * DN];   // 4 KB: rowL | rowR

    const int bid = blockIdx.x;
    const int q = bid & (QN - 1);   // QN = 2^11, fast-varying for L2 locality
    const int b = bid >> 11;

    const float qv = query_t[q];

    // lower_bound over t[1..T-1, b] (length 127): first index with t >= qv.
    // Uniform across the block -> scalar-like, L2-cached loads.
    int lo = 0, hi = TN - 1;
#pragma unroll
    for (int it = 0; it < 7; ++it) {
        if (lo < hi) {
            const int mid = (lo + hi) >> 1;
            const float v = tgrid[(mid + 1) * BN + b];
            if (v < qv) lo = mid + 1; else hi = mid;
        }
    }
    int left = lo;
    if (left > TN - 2) left = TN - 2;

    const float t_left = tgrid[left * BN + b];
    const float t_next = tgrid[(left + 1) * BN + b];
    const float s = (qv - t_left) / (t_next - t_left + 1e-10f);

    // xt[i, b, :] row pointers: i==0 -> x0, i==T-1 -> x1, else knots[b, i-1, :]
    const float* rowL;
    const float* rowR;
    if (left == 0) rowL = x0 + (size_t)b * DN;
    else           rowL = knots + ((size_t)b * (TN - 2) + (size_t)(left - 1)) * DN;
    if (left + 1 == TN - 1) rowR = x1 + (size_t)b * DN;
    else                    rowR = knots + ((size_t)b * (TN - 2) + (size_t)left) * DN;

    const int d = (int)threadIdx.x * 4;   // 128 threads * 4 floats = 512 = D

    // Flat address of a __shared__ object: low 32 bits are the LDS byte offset
    // (LDS aperture mapping, ISA 10.2). Async-load VDST wants that offset.
    const unsigned lds_base = (unsigned)(uintptr_t)smem;
    const unsigned ldsL = lds_base + (unsigned)(d * 4);
    const unsigned ldsR = lds_base + (unsigned)((DN + d) * 4);

    const uint64_t gL = (uint64_t)(uintptr_t)(rowL + d);
    const uint64_t gR = (uint64_t)(uintptr_t)(rowR + d);

    // Per-lane 16B async copies: each wave stages exactly the 1KB (512B per row) it reads back,
    // so a per-wave s_wait_asynccnt is sufficient (no workgroup barrier needed).
    asm volatile("global_load_async_to_lds_b128 %0, %1, off"
                 :: "v"(ldsL), "v"(gL) : "memory");
    asm volatile("global_load_async_to_lds_b128 %0, %1, off"
                 :: "v"(ldsR), "v"(gR) : "memory");
    asm volatile("s_wait_asynccnt 0" ::: "memory");

    const v4f xl = *(const v4f*)(smem + d);
    const v4f xr = *(const v4f*)(smem + DN + d);

    v4f y;
    y.x = __builtin_fmaf(xr.x - xl.x, s, xl.x);
    y.y = __builtin_fmaf(xr.y - xl.y, s, xl.y);
    y.z = __builtin_fmaf(xr.z - xl.z, s, xl.z);
    y.w = __builtin_fmaf(xr.w - xl.w, s, xl.w);

    // Streaming 512MB write: non-temporal so it does not evict the hot knot slab.
    float* o = out + ((size_t)b * QN + (size_t)q) * DN + (size_t)d;
    __builtin_nontemporal_store(y, (v4f*)o);
}

extern "C" void kernel_launch(void* const* d_in, const int* in_sizes, int n_in,
                              void* d_out, int out_size, void* d_ws, size_t ws_size,
                              hipStream_t stream) {
    (void)in_sizes; (void)n_in; (void)out_size; (void)d_ws; (void)ws_size;
    const float* query_t = (const float*)d_in[0];
    const float* knots   = (const float*)d_in[1];
    const float* x0      = (const float*)d_in[2];
    const float* x1      = (const float*)d_in[3];
    const float* tgrid   = (const float*)d_in[4];
    float* out = (float*)d_out;

    dim3 grid(QN * BN);   // 262144 blocks, one output row each
    dim3 block(128);
    hipLaunchKernelGGL(EndPointSpline_18124761989444_kernel, grid, block, 0, stream,
                       query_t, knots, x0, x1, tgrid, out);
}